// Model_42296837931422
// MI455X (gfx1250) — compile-verified
//
#include <hip/hip_runtime.h>
#include <hip/hip_bf16.h>

typedef __attribute__((ext_vector_type(2))) float v2f;
typedef __attribute__((ext_vector_type(8))) float v8f;

#define NN 512           // state dimension
#define NB 128           // batches
#define NGRAPH 32        // graph node count

static __device__ __forceinline__ v8f wmma4(v2f a, v2f b, v8f c) {
    return __builtin_amdgcn_wmma_f32_16x16x4_f32(
        false, a, false, b, (short)0, c, false, false);
}

// ---------------------------------------------------------------------------
// Kernel 1: pack answer bits.  abits[b][w] bit j = (graphs[b, Q[q,0], Q[q,1]] == 1)
// for q = w*32 + j.   graphs/Q arrive as int32 per harness convention.
// ---------------------------------------------------------------------------
__global__ __launch_bounds__(32) void make_abits(const int* __restrict__ graphs,
                                                 const int* __restrict__ Q,
                                                 unsigned* __restrict__ abits) {
    int b = blockIdx.x;
    int t = threadIdx.x;
    if (t < 16) {
        unsigned bits = 0u;
        for (int j = 0; j < 32; ++j) {
            int q = t * 32 + j;
            int r = Q[2 * q + 0];
            int c = Q[2 * q + 1];
            int g = graphs[b * (NGRAPH * NGRAPH) + r * NGRAPH + c];
            bits |= (unsigned)(g == 1) << j;
        }
        abits[b * 16 + t] = bits;
    }
}

// ---------------------------------------------------------------------------
// Kernel 2: row softmax of both logits matrices, stored TRANSPOSED:
//   dstT[m][q] = softmax(src[q,:])[m]
// Transposed layout makes each lane's WMMA B-fragment a contiguous float2.
// ---------------------------------------------------------------------------
__global__ __launch_bounds__(256) void softmax_T(const float* __restrict__ Ln,
                                                 const float* __restrict__ Ly,
                                                 float* __restrict__ PnoT,
                                                 float* __restrict__ PyesT) {
    __shared__ float red[256];
    int r  = blockIdx.x;
    int rr = r & (NN - 1);
    const float* src = (r < NN) ? (Ln + (size_t)rr * NN) : (Ly + (size_t)rr * NN);
    float* dstT      = (r < NN) ? PnoT : PyesT;
    int t = threadIdx.x;

    float x0 = src[t];
    float x1 = src[t + 256];
    float m = fmaxf(x0, x1);
    red[t] = m;
    __syncthreads();
    for (int s = 128; s > 0; s >>= 1) {
        if (t < s) red[t] = fmaxf(red[t], red[t + s]);
        __syncthreads();
    }
    m = red[0];
    __syncthreads();
    float e0 = __expf(x0 - m);
    float e1 = __expf(x1 - m);
    red[t] = e0 + e1;
    __syncthreads();
    for (int s = 128; s > 0; s >>= 1) {
        if (t < s) red[t] += red[t + s];
        __syncthreads();
    }
    float inv = 1.0f / red[0];
    dstT[(size_t)t * NN + rr]         = e0 * inv;
    dstT[(size_t)(t + 256) * NN + rr] = e1 * inv;
}

// ---------------------------------------------------------------------------
// Kernel 3: the 512-step Markov chain.  One workgroup per 16-batch tile,
// 8 waves, each wave owns 64 output columns (4 WMMA n-tiles).
//   S_new = S_no @ Pno + S_yes @ Pyes      via V_WMMA_F32_16X16X4_F32
// State pair lives in LDS (2 * 16*512 f32 = 64 KB).
//
// Inner loop: per-lane VGPR pointers with compile-time immediate offsets
// (one u64 pointer add per pointer per 4-k-tile trip), double-buffered
// fragment registers so loads for tile j+1 overlap WMMAs for tile j.
// ---------------------------------------------------------------------------
__global__ __launch_bounds__(256) void chain_kernel(const float* __restrict__ PnoT,
                                                    const float* __restrict__ PyesT,
                                                    const unsigned* __restrict__ abits_g,
                                                    float* __restrict__ out) {
    __shared__ float s_no[16 * NN];
    __shared__ float s_yes[16 * NN];
    __shared__ unsigned ab[16 * 16];

    const int tid = threadIdx.x;
    const int b0  = blockIdx.x * 16;

    // stage answer bits for this batch tile
    ab[tid] = abits_g[(b0 + (tid >> 4)) * 16 + (tid & 15)];
    // zero state
    for (int i = tid; i < 16 * NN; i += 256) {
        s_no[i]  = 0.0f;
        s_yes[i] = 0.0f;
    }
    __syncthreads();
    // state0 = e0 per batch row, pre-split by answer bit of q=0
    if (tid < 16) {
        unsigned a = ab[tid * 16] & 1u;
        s_no[tid * NN]  = a ? 0.0f : 1.0f;
        s_yes[tid * NN] = a ? 1.0f : 0.0f;
    }
    __syncthreads();

    const int lane  = tid & 31;
    const int wave  = tid >> 5;
    const int lrow  = lane & 15;        // M / N index within tile
    const int halfk = (lane >> 4) * 2;  // lanes 0-15: K={k,k+1}; lanes 16-31: K={k+2,k+3}
    const int rbase = (lane >> 4) * 8;  // D rows: VGPR j -> M = rbase + j
    const int srow  = lrow * NN + halfk;

    int cols[4];
#pragma unroll
    for (int t = 0; t < 4; ++t) cols[t] = wave * 64 + t * 16 + lrow;

    v8f acc[4];
    v2f an0, ay0, bn0[4], by0[4];
    v2f an1, ay1, bn1[4], by1[4];

    for (int step = 0; step < NN; ++step) {
#pragma unroll
        for (int t = 0; t < 4; ++t) acc[t] = (v8f){};

        // per-lane fragment pointers, reset each step
        const v2f* pn[4];
        const v2f* py[4];
#pragma unroll
        for (int t = 0; t < 4; ++t) {
            pn[t] = (const v2f*)&PnoT[cols[t] * NN + halfk];
            py[t] = (const v2f*)&PyesT[cols[t] * NN + halfk];
        }
        int sidx = srow;

        // j = k-tile slot within the current trip (constant -> imm offsets)
        auto load_frag = [&](int j, v2f& an_, v2f& ay_, v2f* bn_, v2f* by_) {
            an_ = *(const v2f*)&s_no[sidx + j * 4];
            ay_ = *(const v2f*)&s_yes[sidx + j * 4];
#pragma unroll
            for (int t = 0; t < 4; ++t) {
                bn_[t] = pn[t][2 * j];
                by_[t] = py[t][2 * j];
            }
        };
        auto compute = [&](v2f an_, v2f ay_, const v2f* bn_, const v2f* by_) {
#pragma unroll
            for (int t = 0; t < 4; ++t) acc[t] = wmma4(an_, bn_[t], acc[t]);
#pragma unroll
            for (int t = 0; t < 4; ++t) acc[t] = wmma4(ay_, by_[t], acc[t]);
        };

        load_frag(0, an0, ay0, bn0, by0);
#pragma unroll 1
        for (int kt = 0; kt < NN / 4; kt += 4) {
            load_frag(1, an1, ay1, bn1, by1);
            compute(an0, ay0, bn0, by0);
            load_frag(2, an0, ay0, bn0, by0);
            compute(an1, ay1, bn1, by1);
            load_frag(3, an1, ay1, bn1, by1);
            compute(an0, ay0, bn0, by0);
            // slot 4 = next trip's slot 0; final-trip overread (8B) lands in
            // adjacent allocated workspace/LDS and the value is dead.
            load_frag(4, an0, ay0, bn0, by0);
            compute(an1, ay1, bn1, by1);
#pragma unroll
            for (int t = 0; t < 4; ++t) { pn[t] += 8; py[t] += 8; }
            sidx += 16;
        }

        __syncthreads();   // all waves done reading old state
#pragma unroll
        for (int t = 0; t < 4; ++t) {
            const int col = cols[t];
#pragma unroll
            for (int j = 0; j < 8; ++j) {
                const int row = rbase + j;
                const unsigned a = (ab[row * 16 + (col >> 5)] >> (col & 31)) & 1u;
                const float v = acc[t][j];
                s_no[row * NN + col]  = a ? 0.0f : v;
                s_yes[row * NN + col] = a ? v : 0.0f;
            }
        }
        __syncthreads();   // new state visible to all waves
    }

    // outputs: state[:,510] then state[:,511]; state = s_no + s_yes
    if (tid < 32) {
        const int i   = tid & 15;
        const int col = (tid < 16) ? (NN - 2) : (NN - 1);
        out[(tid >> 4) * NB + b0 + i] = s_no[i * NN + col] + s_yes[i * NN + col];
    }
}

// ---------------------------------------------------------------------------
extern "C" void kernel_launch(void* const* d_in, const int* in_sizes, int n_in,
                              void* d_out, int out_size, void* d_ws, size_t ws_size,
                              hipStream_t stream) {
    const int*   graphs = (const int*)d_in[0];     // (128,32,32) int
    const int*   Q      = (const int*)d_in[1];     // (512,2) int
    const float* Ln     = (const float*)d_in[2];   // (512,512) f32
    const float* Ly     = (const float*)d_in[3];   // (512,512) f32

    float* PnoT  = (float*)d_ws;                               // 1 MB
    float* PyesT = PnoT + (size_t)NN * NN;                     // 1 MB
    unsigned* abits = (unsigned*)((char*)d_ws + 2u * NN * NN * sizeof(float)); // 8 KB

    make_abits<<<NB, 32, 0, stream>>>(graphs, Q, abits);
    softmax_T<<<2 * NN, 256, 0, stream>>>(Ln, Ly, PnoT, PyesT);
    chain_kernel<<<NB / 16, 256, 0, stream>>>(PnoT, PyesT, abits, (float*)d_out);
}